// SequentialAccumulator_84713934946552
// MI455X (gfx1250) — compile-verified
//
#include <hip/hip_runtime.h>
#include <hip/hip_bf16.h>

// SequentialAccumulator (SpikeFP32) reduce for MI455X / gfx1250.
//
// x:   [8,1024,256,32] f32 bit-planes (0.0/1.0, MSB-first) of [8,1024,256] f32
// out: [8,1024,32]     f32 bit-planes of the strict left-to-right fp32 sum
//                      over axis 2 (order-sensitive, bit-exact).
//
// Roofline: 256 MiB read + 1 MiB write @ 23.3 TB/s  =>  ~11.5 us floor.
// 8192 rows x 255 adds ~ 2 MFLOP: utterly memory bound. WMMA is NOT usable:
// fp32 addition is non-associative and the reference requires the exact
// sequential order, so any tree/matrix reduction changes the result bits.
//
// Strategy (wave32-native):
//   * one wave per output row; lane k owns bit-plane k
//   * per value: one coalesced 128B wave load (nontemporal: data is read once
//     and exceeds the 192MB L2), wave32 ballot assembles the 32 predicate bits
//     into a uint32, s_bfrev flips to MSB-first, bitcast -> exact f32 value
//   * accumulate in strict order; unroll x32 keeps ~4KB of loads in flight
//     per wave to cover HBM latency
//   * encode: lane k extracts bit (31-k) of the result, coalesced 128B store

#define BITS 32
#define NRED 256

__device__ __forceinline__ unsigned int ballot32(bool p) {
#if defined(__AMDGCN__)
  return __builtin_amdgcn_ballot_w32(p);
#else
  (void)p;
  return 0u;
#endif
}

__global__ __launch_bounds__(256) void seqacc_bitplane_kernel(
    const float* __restrict__ x, float* __restrict__ out, int rows)
{
  const int lane = (int)(threadIdx.x & 31u);
  const int row  = (int)((blockIdx.x * blockDim.x + threadIdx.x) >> 5);
  if (row >= rows) return;

  // Row base: lane k reads bit-plane k of each value -> coalesced 128B/wave.
  const float* __restrict__ p = x + (size_t)row * (NRED * BITS) + lane;

  // n = 0 seeds the accumulator (carry = seq[0] in the reference scan);
  // do NOT start from 0.0f (0.0 + (-0.0) != -0.0 bit-exactly).
  float b0 = __builtin_nontemporal_load(p);
  float acc = __uint_as_float(__brev(ballot32(b0 != 0.0f)));

  #pragma unroll 32
  for (int n = 1; n < NRED; ++n) {
    float b = __builtin_nontemporal_load(p + (size_t)n * BITS);
    unsigned int u = __brev(ballot32(b != 0.0f));  // exact IEEE-754 bits
    acc += __uint_as_float(u);                     // strict order: carry + v_n
  }

  // Re-encode: out[row, j] = (bits(acc) >> (31 - j)) & 1, j = lane.
  const unsigned int u = __float_as_uint(acc);
  out[(size_t)row * BITS + lane] = (float)((u >> (31 - lane)) & 1u);
}

extern "C" void kernel_launch(void* const* d_in, const int* in_sizes, int n_in,
                              void* d_out, int out_size, void* d_ws, size_t ws_size,
                              hipStream_t stream) {
  (void)n_in; (void)out_size; (void)d_ws; (void)ws_size;

  const float* x = (const float*)d_in[0];   // [rows, 256, 32] flat
  // d_in[1] is the python scalar `dim` (== 2); layout is fixed, ignore it.
  float* out = (float*)d_out;               // [rows, 32] flat

  const int rows = in_sizes[0] / (NRED * BITS);   // 8*1024 = 8192

  const int wavesPerBlock = 8;                    // 256 threads = 8 wave32
  const int threads = wavesPerBlock * 32;
  const int blocks  = (rows + wavesPerBlock - 1) / wavesPerBlock;

  seqacc_bitplane_kernel<<<blocks, threads, 0, stream>>>(x, out, rows);
}